// ISTFT_4922032522000
// MI455X (gfx1250) — compile-verified
//
#include <hip/hip_runtime.h>
#include <math.h>

typedef float v2f __attribute__((ext_vector_type(2)));
typedef float v8f __attribute__((ext_vector_type(8)));

namespace {
constexpr int kNfft   = 400;           // 2*(F-1)
constexpr int kFreq   = 201;
constexpr int kK      = 402;           // re/im interleaved K dimension
constexpr int kKpad   = 404;           // K padded to multiple of 4 (zero rows)
constexpr int kHop    = 160;
constexpr int kT      = 3000;
constexpr int kBatch  = 16;
constexpr int kFrames = kBatch * kT;   // 48000
constexpr int kOutLen = 479840;        // est_len - 2*(n_fft/2)
constexpr int kHalfN  = kNfft / 2;     // 200 (trim)
constexpr double kTwoPi = 6.283185307179586476925286766559;
}

// ---------------------------------------------------------------------------
// Zero the output (harness poisons it with 0xAA).
// ---------------------------------------------------------------------------
__global__ void zero_out_kernel(float* __restrict__ out, int n) {
    int i = blockIdx.x * blockDim.x + threadIdx.x;
    if (i < n) out[i] = 0.0f;
}

// ---------------------------------------------------------------------------
// Build windowed irfft basis, layout basis[k][n], k in [0,404), n in [0,400).
//   k = 2j   : Re coeff:  win/N        (j==0)
//                         2*win/N*cos  (1<=j<=199)
//                         win/N*(-1)^n (j==200, Nyquist)
//   k = 2j+1 : Im coeff: -2*win/N*sin  (1<=j<=199), else 0 (irfft drops
//                         Im of DC and Nyquist bins)
//   k >= 402 : 0 (padding)
// ---------------------------------------------------------------------------
__global__ void build_basis_kernel(float* __restrict__ basis) {
    int idx = blockIdx.x * blockDim.x + threadIdx.x;
    if (idx >= kKpad * kNfft) return;
    int k = idx / kNfft;
    int n = idx - k * kNfft;
    double v = 0.0;
    if (k < kK) {
        double win = 0.54 - 0.46 * cos(kTwoPi * (double)n / (double)kNfft);
        double s   = win / (double)kNfft;
        int j = k >> 1;
        if ((k & 1) == 0) {
            if (j == 0)                 v = s;
            else if (j == kFreq - 1)    v = (n & 1) ? -s : s;
            else                        v = 2.0 * s * cos(kTwoPi * (double)(j * n) / (double)kNfft);
        } else {
            if (j >= 1 && j <= kFreq - 2)
                v = -2.0 * s * sin(kTwoPi * (double)(j * n) / (double)kNfft);
        }
    }
    basis[idx] = (float)v;
}

// ---------------------------------------------------------------------------
// Core GEMM: q[f][n] = sum_k x[f][k] * basis[k][n] using V_WMMA_F32_16X16X4_F32,
// fused with the overlap-add scatter (atomic f32 adds; <=3 colliders/sample).
// Grid: 3000 frame-tiles (16 frames each). Block: 160 threads = 5 waves;
// wave w owns n-tiles [5w, 5w+5) of the 25 tiles covering n in [0,400).
// K loop is tail-peeled so the 100-pair main loop is branch-free.
// ---------------------------------------------------------------------------
__global__ void __launch_bounds__(160)
istft_gemm_kernel(const float* __restrict__ x,
                  const float* __restrict__ basis,
                  float* __restrict__ out) {
    const int ftile = blockIdx.x;            // 0..2999
    const int wave  = threadIdx.x >> 5;      // 0..4
    const int lane  = threadIdx.x & 31;
    const int half  = lane >> 4;             // 0: K pair (k0,k0+1), 1: (k0+2,k0+3)
    const int l15   = lane & 15;
    const int nbase = wave * 80 + l15;       // first n column for this lane

    // A fragment source: lane l15 holds row (frame) ftile*16 + l15.
    // Pointer pre-offset by this lane's K sub-pair.
    const float* arow = x + (size_t)(ftile * 16 + l15) * kK + 2 * half;
    const float* bcol = basis + (size_t)(2 * half) * kNfft + nbase;

    v8f acc[5];
#pragma unroll
    for (int nt = 0; nt < 5; ++nt) acc[nt] = (v8f){};

    // ---- main K loop: k = 0..399, two K-steps (8 values) per iteration ----
    for (int k0 = 0; k0 < kNfft; k0 += 8) {
        const v2f a0 = *(const v2f*)(arow + k0);
        const v2f a1 = *(const v2f*)(arow + k0 + 4);
        const float* b0 = bcol + (size_t)k0 * kNfft;
        const float* b1 = b0 + (size_t)4 * kNfft;
        v2f bv0[5], bv1[5];
#pragma unroll
        for (int nt = 0; nt < 5; ++nt) {
            bv0[nt].x = b0[nt * 16];
            bv0[nt].y = b0[kNfft + nt * 16];
            bv1[nt].x = b1[nt * 16];
            bv1[nt].y = b1[kNfft + nt * 16];
        }
#pragma unroll
        for (int nt = 0; nt < 5; ++nt)
            acc[nt] = __builtin_amdgcn_wmma_f32_16x16x4_f32(
                false, a0, false, bv0[nt], (short)0, acc[nt], false, false);
#pragma unroll
        for (int nt = 0; nt < 5; ++nt)
            acc[nt] = __builtin_amdgcn_wmma_f32_16x16x4_f32(
                false, a1, false, bv1[nt], (short)0, acc[nt], false, false);
    }

    // ---- K tail: k = 400..403 (only k=400, the real Nyquist, is nonzero) ---
    {
        // half=0 lanes need A[f][400..401]; half=1 lanes would read 402..403
        // (OOB on the last frame) -> clamp their address in-bounds, then zero.
        v2f a = *(const v2f*)(arow + (half ? -2 : kNfft));
        if (half) a = (v2f){};               // cndmask, no exec divergence issue
        const float* brow = bcol + (size_t)kNfft * kNfft;  // rows 400..403 (401+ zero)
#pragma unroll
        for (int nt = 0; nt < 5; ++nt) {
            v2f b;
            b.x = brow[nt * 16];
            b.y = brow[kNfft + nt * 16];
            acc[nt] = __builtin_amdgcn_wmma_f32_16x16x4_f32(
                false, a, false, b, (short)0, acc[nt], false, false);
        }
    }

    // D layout: VGPR i, lanes<16 -> M=i, N=lane; lanes>=16 -> M=i+8, N=lane-16.
#pragma unroll
    for (int nt = 0; nt < 5; ++nt) {
        const int n = nbase + nt * 16;
#pragma unroll
        for (int i = 0; i < 8; ++i) {
            const int fr = ftile * 16 + half * 8 + i;   // global frame index
            const int b  = fr / kT;
            const int t  = fr - b * kT;
            const int p  = t * kHop + n - kHalfN;       // trimmed output position
            if (p >= 0 && p < kOutLen) {
                atomicAdd(out + (size_t)b * kOutLen + p, acc[nt][i]);
            }
        }
    }
}

// ---------------------------------------------------------------------------
// Fallback (only if ws can't hold the 646KB basis): direct per-sample DFT.
// ---------------------------------------------------------------------------
__global__ void istft_naive_kernel(const float* __restrict__ x,
                                   float* __restrict__ out) {
    long idx = (long)blockIdx.x * blockDim.x + threadIdx.x;
    if (idx >= (long)kFrames * kNfft) return;
    int f = (int)(idx / kNfft);
    int n = (int)(idx - (long)f * kNfft);
    const float* a = x + (size_t)f * kK;
    float acc = a[0] + ((n & 1) ? -a[2 * (kFreq - 1)] : a[2 * (kFreq - 1)]);
    for (int j = 1; j < kFreq - 1; ++j) {
        int m = (j * n) % kNfft;             // exact angle reduction
        float s, c;
        __sincosf((float)(kTwoPi * m / kNfft), &s, &c);
        acc += 2.0f * (a[2 * j] * c - a[2 * j + 1] * s);
    }
    float win = 0.54f - 0.46f * __cosf((float)(kTwoPi * n / kNfft));
    float q = acc * win * (1.0f / kNfft);
    int b = f / kT, t = f - b * kT;
    int p = t * kHop + n - kHalfN;
    if (p >= 0 && p < kOutLen) atomicAdd(out + (size_t)b * kOutLen + p, q);
}

// ---------------------------------------------------------------------------
// Normalize by sum of squared windows (computed analytically, <=3 terms) and
// apply the ssw > eps guard, in place.
// ---------------------------------------------------------------------------
__global__ void normalize_kernel(float* __restrict__ out, int total) {
    int idx = blockIdx.x * blockDim.x + threadIdx.x;
    if (idx >= total) return;
    int b  = idx / kOutLen;
    int po = idx - b * kOutLen;
    int p  = po + kHalfN;                    // position in untrimmed est_len frame
    int t_hi = p / kHop;            if (t_hi > kT - 1) t_hi = kT - 1;
    int t_lo = (p - (kNfft - 1) + kHop - 1) / kHop;
    if (t_lo < 0) t_lo = 0;
    float ssw = 0.0f;
    for (int t = t_lo; t <= t_hi; ++t) {
        int n = p - t * kHop;                // 0..399 by construction
        float w = 0.54f - 0.46f * __cosf((float)(kTwoPi * n / kNfft));
        ssw += w * w;
    }
    float v = out[idx];
    out[idx] = (ssw > 1e-12f) ? (v / ssw) : v;
}

// ---------------------------------------------------------------------------
extern "C" void kernel_launch(void* const* d_in, const int* in_sizes, int n_in,
                              void* d_out, int out_size, void* d_ws, size_t ws_size,
                              hipStream_t stream) {
    const float* x = (const float*)d_in[0];   // (16, 3000, 201, 2) fp32
    float* out = (float*)d_out;               // (16, 479840) fp32

    {
        int thr = 256;
        int blk = (out_size + thr - 1) / thr;
        zero_out_kernel<<<blk, thr, 0, stream>>>(out, out_size);
    }

    const size_t basisBytes = (size_t)kKpad * kNfft * sizeof(float); // 646,400 B
    if (ws_size >= basisBytes) {
        float* basis = (float*)d_ws;
        int tot = kKpad * kNfft;
        build_basis_kernel<<<(tot + 255) / 256, 256, 0, stream>>>(basis);
        istft_gemm_kernel<<<kFrames / 16, 160, 0, stream>>>(x, basis, out);
    } else {
        long tot = (long)kFrames * kNfft;
        istft_naive_kernel<<<(int)((tot + 255) / 256), 256, 0, stream>>>(x, out);
    }

    {
        int thr = 256;
        int blk = (out_size + thr - 1) / thr;
        normalize_kernel<<<blk, thr, 0, stream>>>(out, out_size);
    }
}